// AdderConvReLUBlock_8693013807225
// MI455X (gfx1250) — compile-verified
//
#include <hip/hip_runtime.h>
#include <stdint.h>

#define BATCH 4
#define C 32          // Cin == Cout
#define H 128
#define W 128
#define KS 3
#define TH 8          // output tile height
#define TW 16         // output tile width
#define XT_H (TH + 2) // 10
#define XT_W (TW + 2) // 18
#define XT_SZ (C * XT_H * XT_W)  // 5760 floats = 22.5 KB
#define WL_SZ (KS * KS * C * C)  // 9216 floats = 36 KB
#define NTHREADS 128

typedef __attribute__((ext_vector_type(4))) unsigned int u32x4;
typedef __attribute__((ext_vector_type(8))) int i32x8;
typedef __attribute__((ext_vector_type(4))) int i32x4;

__global__ __launch_bounds__(NTHREADS) void adder_conv_relu_kernel(
    const float* __restrict__ x, const float* __restrict__ w,
    float* __restrict__ out)
{
    __shared__ __align__(16) float xs[XT_SZ];   // [ci][y][x], contiguous (TDM fill order)
    __shared__ __align__(16) float wl[WL_SZ];   // [tap][ci][co]

    const int tid  = threadIdx.x;
    const int b    = blockIdx.y;
    const int tile = blockIdx.x;      // 0..127
    const int ct   = tile & 7;        // 0..7   (col tile)
    const int rt   = tile >> 3;       // 0..15  (row tile)
    const int row0 = rt * TH;
    const int col0 = ct * TW;

    // ---- Issue TDM first so the DMA overlaps the weight-staging loop ----
    // Tensor Data Mover: 3D tile X=XT_W, Y=XT_H, Z=C channels.
    // Tile origin at (row0-1, col0-1); right/bottom overhang relies on
    // TDM OOB-read -> zero, matching the reference zero padding.
    const bool interior = (rt > 0) && (ct > 0);
    if (interior && tid < warpSize) {  // wave 0 issues the DMA (EXEC-independent op)
        unsigned long long ga = (unsigned long long)(uintptr_t)x
            + 4ull * ((unsigned long long)b * (C * H * W)
                      + (unsigned long long)(row0 - 1) * W
                      + (unsigned long long)(col0 - 1));
        unsigned lds_off = (unsigned)(uintptr_t)(&xs[0]);

        u32x4 g0;
        g0[0] = 1u;                         // count=1, user descriptor
        g0[1] = lds_off;                    // lds_addr (bytes)
        g0[2] = (unsigned)ga;               // global_addr[31:0]
        g0[3] = (unsigned)((ga >> 32) & 0x1FFFFFFull) | (2u << 30); // type=2

        i32x8 g1;
        g1[0] = (int)(2u << 16);                               // data_size=4B
        g1[1] = (int)(((unsigned)W & 0xFFFFu) << 16);          // tensor_dim0 lo16
        g1[2] = (int)((((unsigned)W >> 16) & 0xFFFFu)
                      | (((unsigned)H & 0xFFFFu) << 16));      // dim0 hi | dim1 lo
        g1[3] = (int)((((unsigned)H >> 16) & 0xFFFFu)
                      | ((unsigned)XT_W << 16));               // dim1 hi | tile_dim0
        g1[4] = (int)((unsigned)XT_H | ((unsigned)C << 16));   // tile_dim1 | tile_dim2
        g1[5] = (int)(unsigned)W;                              // dim0_stride lo32 (row)
        g1[6] = (int)((((unsigned)(H * W)) & 0xFFFFu) << 16);  // stride0 hi16 | stride1 lo16
        g1[7] = (int)(((unsigned)(H * W)) >> 16);              // stride1 bits 47:16

        i32x4 g2 = {(int)C, 0, 0, 0};   // tensor_dim2 = channels; dim3/tile_dim3 unused
        i32x4 g3 = {0, 0, 0, 0};

#if __clang_major__ >= 23
        i32x8 g4 = {0, 0, 0, 0, 0, 0, 0, 0};
        __builtin_amdgcn_tensor_load_to_lds(g0, g1, g2, g3, g4, 0);
#else
        __builtin_amdgcn_tensor_load_to_lds(g0, g1, g2, g3, 0);
#endif
    }

    // ---- Stage weights into LDS, reordered to [tap][ci][co] (all threads),
    //      concurrent with the in-flight TDM transfer ----
    for (int i = tid; i < WL_SZ; i += NTHREADS) {
        int tap = i >> 10;        // i / 1024
        int r   = i & 1023;
        int ci  = r >> 5;
        int co  = r & 31;
        wl[i] = w[co * (C * KS * KS) + ci * (KS * KS) + tap];
    }

    // ---- Finish x staging ----
    if (interior) {
        if (tid < warpSize)
            __builtin_amdgcn_s_wait_tensorcnt((short)0);
    } else {
        // Top/left-edge tiles: guarded loads (halo may precede the tensor).
        for (int i = tid; i < XT_SZ; i += NTHREADS) {
            int ci = i / (XT_H * XT_W);
            int r  = i - ci * (XT_H * XT_W);
            int y  = r / XT_W;
            int xc = r - y * XT_W;
            int gy = row0 - 1 + y;
            int gx = col0 - 1 + xc;
            float v = 0.0f;
            if ((unsigned)gy < (unsigned)H && (unsigned)gx < (unsigned)W)
                v = x[((b * C + ci) * H + gy) * W + gx];
            xs[i] = v;
        }
    }
    __syncthreads();

    // ---- Compute: thread = one pixel, 32 cout accumulators in VGPRs ----
    const int px = tid & (TW - 1);
    const int py = tid >> 4;

    float acc[C];
#pragma unroll
    for (int co = 0; co < C; ++co) acc[co] = 0.0f;

    for (int ci = 0; ci < C; ++ci) {
        const float* xrow = &xs[ci * (XT_H * XT_W)];
#pragma unroll
        for (int ky = 0; ky < KS; ++ky) {
#pragma unroll
            for (int kx = 0; kx < KS; ++kx) {
                float xv = xrow[(py + ky) * XT_W + (px + kx)];
                const float* wrow = &wl[(ky * KS + kx) * (C * C) + ci * C];
                // 32 consecutive floats -> broadcast ds_load_b128 x8;
                // each xv feeds 64 VALU ops (v_sub_f32 + v_add_f32 with |abs|).
#pragma unroll
                for (int co = 0; co < C; ++co) {
                    acc[co] += __builtin_fabsf(xv - wrow[co]);
                }
            }
        }
    }

    // ---- ReLU(-acc) and coalesced stores ----
    const int orow = row0 + py;
    const int ocol = col0 + px;
#pragma unroll
    for (int co = 0; co < C; ++co) {
        float v = -acc[co];
        out[((b * C + co) * H + orow) * W + ocol] = v > 0.0f ? v : 0.0f;
    }
}

extern "C" void kernel_launch(void* const* d_in, const int* in_sizes, int n_in,
                              void* d_out, int out_size, void* d_ws, size_t ws_size,
                              hipStream_t stream) {
    const float* x = (const float*)d_in[0];   // (4,32,128,128) f32
    const float* w = (const float*)d_in[1];   // (32,32,3,3) f32
    float* out = (float*)d_out;               // (4,32,128,128) f32
    (void)in_sizes; (void)n_in; (void)out_size; (void)d_ws; (void)ws_size;

    dim3 grid(128, BATCH);   // 16x8 tiles per image, 4 images
    dim3 block(NTHREADS);
    adder_conv_relu_kernel<<<grid, block, 0, stream>>>(x, w, out);
}